// flowTracker_17033840296558
// MI455X (gfx1250) — compile-verified
//
#include <hip/hip_runtime.h>
#include <stdint.h>

// ROI adaptive-max-pool (7x7) + FC(98->4) + ReLU + box add, CDNA5/gfx1250.
// 8 blocks x 256 threads (8 wave32), 16 boxes per block.
// Feature map (120 KB) staged into LDS via the Tensor Data Mover (TDM) when
// the builtin is available (falls back to b128 copies); FC tile computed with
// V_WMMA_F32_16X16X4_F32 from branch-free ds_load_b64 fragments.

#define PP 7
#define CH 2
#define FH 96
#define FW 160
#define FELEMS (CH * FH * FW)   // 30720 floats
#define KDIM 98                 // CH * 7 * 7
#define LDK 104                 // padded row stride (bank spread, 8B aligned)
#define BOXES_PER_BLOCK 16

typedef float    v2f  __attribute__((ext_vector_type(2)));
typedef float    v8f  __attribute__((ext_vector_type(8)));
typedef uint32_t u32x4 __attribute__((ext_vector_type(4)));
typedef int      i32x4 __attribute__((ext_vector_type(4)));
typedef int      i32x8 __attribute__((ext_vector_type(8)));

#define HAS_TDM (__has_builtin(__builtin_amdgcn_tensor_load_to_lds) && \
                 __has_builtin(__builtin_amdgcn_s_wait_tensorcnt))

__global__ __launch_bounds__(256, 1)
void roi_pool_fc_kernel(const float* __restrict__ feat,
                        const float* __restrict__ boxes,
                        const float* __restrict__ fcw,
                        const float* __restrict__ fcb,
                        float* __restrict__ out)
{
    __shared__ float sFeat[FELEMS];                  // 120 KB
    __shared__ float sPool[BOXES_PER_BLOCK * LDK];   // 6.5 KB, A-tile (padded)
    __shared__ float sW[16 * LDK];                   // 6.5 KB, B-tile (padded)

    const int tid = threadIdx.x;

    // ---- Stage 0a: feature map -> LDS ----
#if HAS_TDM
    if (tid < 32) {
        // Tensor DMA descriptor (D#), CDNA5 ISA.
        // 2-D tensor: one row of 30720 fp32 elements; tile == tensor.
        const uint64_t ga  = (uint64_t)(uintptr_t)feat;        // global byte addr
        const uint32_t lds = (uint32_t)(uintptr_t)&sFeat[0];   // LDS byte offset

        u32x4 g0;
        g0[0] = 1u;                                   // count=1, user mode
        g0[1] = lds;                                  // lds_addr
        g0[2] = (uint32_t)ga;                         // global_addr[31:0]
        g0[3] = (uint32_t)((ga >> 32) & 0x1FFFFFFu)   // global_addr[56:32]
              | (2u << 30);                           // type = 2 ("image")

        i32x8 g1;
        g1[0] = (int)(2u << 16);                      // data_size = 4 bytes
        g1[1] = (int)((FELEMS & 0xFFFFu) << 16);      // tensor_dim0[15:0]
        g1[2] = (int)((FELEMS >> 16) | (1u << 16));   // dim0[31:16] | tensor_dim1=1
        g1[3] = (int)((FELEMS & 0xFFFFu) << 16);      // tile_dim0 = 30720
        g1[4] = 1;                                    // tile_dim1 = 1
        g1[5] = (int)FELEMS;                          // tensor_dim0_stride[31:0]
        g1[6] = (int)((FELEMS & 0xFFFFu) << 16);      // stride0[47:32]=0 | stride1 lo
        g1[7] = 0;

        const i32x4 z4 = {0, 0, 0, 0};
#if __clang_major__ >= 23
        const i32x8 z8 = {0, 0, 0, 0, 0, 0, 0, 0};
        __builtin_amdgcn_tensor_load_to_lds(g0, g1, z4, z4, z8, 0);
#else
        __builtin_amdgcn_tensor_load_to_lds(g0, g1, z4, z4, 0);
#endif
    }
#else
    {
        const float4* src = reinterpret_cast<const float4*>(feat);
        float4*       dst = reinterpret_cast<float4*>(sFeat);
        for (int i = tid; i < FELEMS / 4; i += 256)
            dst[i] = src[i];
    }
#endif

    // ---- Stage 0b: zero-fill padded B tile and A-tile pad columns ----
    for (int i = tid; i < 16 * LDK; i += 256) sW[i] = 0.0f;
    if (tid < BOXES_PER_BLOCK * (LDK - KDIM)) {       // 16 boxes x cols 98..103
        const int b = tid / (LDK - KDIM);
        const int k = KDIM + tid % (LDK - KDIM);
        sPool[b * LDK + k] = 0.0f;
    }
    if (tid < 32) {
        __builtin_prefetch(fcw + tid * 4, 0, 0);      // global_prefetch_b8
#if HAS_TDM
        __builtin_amdgcn_s_wait_tensorcnt(0);         // drain TDM (wave 0 issued)
#endif
    }
    __syncthreads();

    // ---- Stage 1: pooling (16 boxes x 98 cells) + weight staging ----
    const int boxBase = blockIdx.x * BOXES_PER_BLOCK;
    for (int t = tid; t < BOXES_PER_BLOCK * KDIM; t += 256) {
        const int bl   = t / KDIM;
        const int cell = t - bl * KDIM;               // c*49 + i*7 + j
        const int c    = cell / 49;
        const int ij   = cell - c * 49;
        const int i    = ij / 7;
        const int j    = ij - i * 7;
        const int gb   = boxBase + bl;

        const int x1 = (int)boxes[4 * gb + 0];
        const int y1 = (int)boxes[4 * gb + 1];
        const int x2 = (int)boxes[4 * gb + 2];
        const int y2 = (int)boxes[4 * gb + 3];
        const int hh = y2 - y1, ww = x2 - x1;

        const int r0 = y1 + (i * hh) / PP;
        const int r1 = y1 + ((i + 1) * hh + PP - 1) / PP;
        const int q0 = x1 + (j * ww) / PP;
        const int q1 = x1 + ((j + 1) * ww + PP - 1) / PP;

        float m = -__builtin_inff();
        const float* base = sFeat + c * (FH * FW);
        for (int r = r0; r < r1; ++r) {
            const float* rowp = base + r * FW;
            for (int q = q0; q < q1; ++q)
                m = fmaxf(m, rowp[q]);
        }
        sPool[bl * LDK + cell] = m;
    }
    // fc_w (4 x 98) -> sW[n*LDK + k]; rows 4..15 / cols 98..103 stay zero
    for (int t = tid; t < 4 * KDIM; t += 256)
        sW[(t / KDIM) * LDK + (t % KDIM)] = fcw[t];
    __syncthreads();

    // ---- Stage 2: FC tile via WMMA (wave 0; EXEC all-1s in-wave) ----
    if (tid < 32) {
        const int lane = tid;
        const int half = lane >> 4;      // 0: lanes 0-15, 1: lanes 16-31
        const int mrow = lane & 15;      // A-matrix row this lane supplies
        const int ncol = lane & 15;      // B/D column this lane holds

        // Per-lane 8B-aligned fragment base offsets (k0 added per step).
        const float* aBase = &sPool[mrow * LDK + 2 * half];
        const float* bBase = &sW[ncol * LDK + 2 * half];

        v8f acc = {};
        #pragma unroll
        for (int k0 = 0; k0 < 100; k0 += 4) {         // K padded 98 -> 100
            const v2f a = *reinterpret_cast<const v2f*>(aBase + k0); // ds_load_b64
            const v2f b = *reinterpret_cast<const v2f*>(bBase + k0); // ds_load_b64
            acc = __builtin_amdgcn_wmma_f32_16x16x4_f32(
                false, a, false, b, (short)0, acc, false, false);
        }

        // ---- Epilogue: relu(acc + bias) + boxes; only cols 0..3 valid ----
        if (ncol < 4) {
            const float bias = fcb[ncol];
            #pragma unroll
            for (int v = 0; v < 8; ++v) {
                const int m  = v + 8 * half;          // D row: VGPR v, lane-half
                const int gb = boxBase + m;
                const float d = fmaxf(acc[v] + bias, 0.0f);
                out[4 * gb + ncol] = d + boxes[4 * gb + ncol];
            }
        }
    }
}

extern "C" void kernel_launch(void* const* d_in, const int* in_sizes, int n_in,
                              void* d_out, int out_size, void* d_ws, size_t ws_size,
                              hipStream_t stream) {
    (void)n_in; (void)out_size; (void)d_ws; (void)ws_size;
    const float* feat  = (const float*)d_in[0];   // [1,2,96,160]
    const float* boxes = (const float*)d_in[1];   // [N,4]
    const float* fcw   = (const float*)d_in[2];   // [4, 98]
    const float* fcb   = (const float*)d_in[3];   // [4]
    float* out = (float*)d_out;                   // [N,4]

    const int nBoxes = in_sizes[1] / 4;                                   // 128
    const int grid   = (nBoxes + BOXES_PER_BLOCK - 1) / BOXES_PER_BLOCK;  // 8
    roi_pool_fc_kernel<<<grid, 256, 0, stream>>>(feat, boxes, fcw, fcb, out);
}